// ClusterSwinEncoder_50457275793556
// MI455X (gfx1250) — compile-verified
//
#include <hip/hip_runtime.h>
#include <hip/hip_bf16.h>
#include <stdint.h>

// ---------------------------------------------------------------------------
// ClusterSwinEncoder forward for MI455X (gfx1250, wave32, WMMA + async LDS).
//
//   1) proj_gemm_wmma<C,HW> : 1x1-conv GEMMs via V_WMMA_F32_16X16X4_F32.
//      - X tile staged to LDS with global_load_async_to_lds_b32 (ASYNCcnt),
//        double-buffered K=64 chunks overlapped with compute.
//      - LDS layout is K-pair interleaved: element (c,p) at dword
//        (c>>1)*96 + p*2 + (c&1), so each B fragment {K,K+1} is one
//        contiguous 8B word -> single ds_load_b64, even-aligned operand,
//        and the 96-dword pair stride ( == 32 mod 64 banks ) makes the two
//        half-wave K groups hit disjoint bank halves (conflict-free).
//      - Each wave computes a 32(O) x 32(P) tile: 4 accumulators,
//        4 WMMAs per 2 global b64 + 2 ds b64.
//   2) upsample_add : fmap += bilinear_up(f3p) + bilinear_up(f4p)
//   3) reduce_g     : g[n,o] = mean_hw fmap   (one wave32 per row)
//   4) heat_kernel  : heat[n,p] = mean_c |fmap|
//   5) topk_gather  : per-image top-8 argmax + channel-vector gather
// ---------------------------------------------------------------------------

typedef float v2f __attribute__((ext_vector_type(2)));
typedef float v8f __attribute__((ext_vector_type(8)));

#define EMBED   512
#define HW2     784   // 28*28
#define HW3     196   // 14*14
#define HW4     49    // 7*7
#define NEG_INF (-3.402823466e38f)

// ---------------------------------------------------------------------------
// GEMM: Y[n,o,p] = bias[o] + sum_c Wm[o,c] * X[n,c,p]
//
// A (16x4 f32): lane L<16 holds (M=L, K=0/1), lane L>=16 holds (M=L-16, K=2/3)
// B (4x16 f32): lanes 0-15 hold K-rows 0/1, lanes 16-31 hold K-rows 2/3
// D (16x16 f32): VGPR r -> row M = r + (lane>=16 ? 8 : 0), col N = lane&15
// ---------------------------------------------------------------------------
template<int C, int HW>
__global__ __launch_bounds__(256) void proj_gemm_wmma(
    const float* __restrict__ X,    // [N, C, HW]
    const float* __restrict__ Wm,   // [EMBED, C]
    const float* __restrict__ bias, // [EMBED]
    float* __restrict__ Y)          // [N, EMBED, HW]
{
    constexpr int KCH = 64;         // K-chunk per double-buffer slot
    constexpr int PS  = 96;         // dword stride between K-pair rows
    __shared__ __align__(16) float ldsX[2][(KCH / 2) * PS];

    const int lane  = threadIdx.x & 31;
    const int wave  = threadIdx.x >> 5;            // 0..7
    const int n     = blockIdx.z;
    const int pBase = blockIdx.x * 32;
    const int oBase = blockIdx.y * 256 + wave * 32;

    const int laneM = lane & 15;
    const int kHi   = (lane >> 4) << 1;            // 0 or 2

    const float* __restrict__ xn   = X + (size_t)n * C * HW;
    const float* __restrict__ wrow = Wm + (size_t)(oBase + laneM) * C + kHi;

    // --- cooperative async staging: chunk = 64 K-rows x 32 P-cols -----------
    const int sCol  = threadIdx.x & 31;            // P column
    const int sRow0 = threadIdx.x >> 5;            // K rows sRow0 + 8*i
    int pg = pBase + sCol;                         // clamp for OOB-safe loads
    if (pg > HW - 1) pg = HW - 1;
    // interleaved destination: (c>>1)*PS + p*2 + (c&1), c = sRow0 + 8*i
    const uint32_t dstBase =
        (uint32_t)(((sRow0 >> 1) * PS + sCol * 2 + (sRow0 & 1)) * 4);

    auto stage = [&](int kc, int buf) {
        const float* src = xn + (size_t)(kc + sRow0) * HW + pg;
        uint32_t dst = (uint32_t)(uintptr_t)&ldsX[buf][0] + dstBase;
        #pragma unroll
        for (int i = 0; i < KCH / 8; ++i) {
            asm volatile("global_load_async_to_lds_b32 %0, %1, off"
                         :: "v"(dst), "v"(src) : "memory");
            dst += 4 * PS * 4;      // 4 pair-rows per 8 c-rows
            src += (size_t)8 * HW;
        }
    };

    stage(0, 0);
    asm volatile("s_wait_asynccnt 0x0" ::: "memory");
    __syncthreads();

    v8f acc00 = {}, acc01 = {}, acc10 = {}, acc11 = {};

    for (int kc = 0; kc < C; kc += KCH) {
        const int cur = (kc / KCH) & 1;
        if (kc + KCH < C) stage(kc + KCH, cur ^ 1);

        const float* __restrict__ wb = wrow + kc;
        // per-wave B base: pair-row (kHi>>1), column laneM (tile0)
        const float* __restrict__ lb =
            &ldsX[cur][(kHi >> 1) * PS + laneM * 2];
        #pragma unroll
        for (int kk = 0; kk < KCH; kk += 4) {
            v2f a0, a1, b0, b1;
            const float* wp = wb + kk;
            a0.x = wp[0];
            a0.y = wp[1];
            a1.x = wp[16 * C];
            a1.y = wp[16 * C + 1];
            const float* bp = lb + (kk >> 1) * PS;
            b0 = *(const v2f*)(bp);        // {K,K+1} for cols 0..15
            b1 = *(const v2f*)(bp + 32);   // {K,K+1} for cols 16..31
            acc00 = __builtin_amdgcn_wmma_f32_16x16x4_f32(
                false, a0, false, b0, (short)0, acc00, false, false);
            acc01 = __builtin_amdgcn_wmma_f32_16x16x4_f32(
                false, a0, false, b1, (short)0, acc01, false, false);
            acc10 = __builtin_amdgcn_wmma_f32_16x16x4_f32(
                false, a1, false, b0, (short)0, acc10, false, false);
            acc11 = __builtin_amdgcn_wmma_f32_16x16x4_f32(
                false, a1, false, b1, (short)0, acc11, false, false);
        }

        asm volatile("s_wait_asynccnt 0x0" ::: "memory");
        __syncthreads();
    }

    // --- store (mask OOB columns only; WMMA ran with full EXEC) -------------
    const int pc0  = pBase + laneM;
    const int pc1  = pBase + 16 + laneM;
    const int oSub = (lane >> 4) << 3;             // 0 or 8
    float* __restrict__ yo = Y + (size_t)n * EMBED * HW;
    #pragma unroll
    for (int r = 0; r < 8; ++r) {
        const int o0 = oBase + oSub + r;
        const int o1 = o0 + 16;
        const float bz0 = bias[o0];
        const float bz1 = bias[o1];
        if (pc0 < HW) {
            yo[(size_t)o0 * HW + pc0] = acc00[r] + bz0;
            yo[(size_t)o1 * HW + pc0] = acc10[r] + bz1;
        }
        if (pc1 < HW) {
            yo[(size_t)o0 * HW + pc1] = acc01[r] + bz0;
            yo[(size_t)o1 * HW + pc1] = acc11[r] + bz1;
        }
    }
}

// ---------------------------------------------------------------------------
// Half-pixel bilinear sample (align_corners=False), clamped edges.
// ---------------------------------------------------------------------------
__device__ __forceinline__ float bil(const float* __restrict__ img, int S,
                                     int h, int w)
{
    const float scale = (float)S * (1.0f / 28.0f);
    float sy = (h + 0.5f) * scale - 0.5f;
    float sx = (w + 0.5f) * scale - 0.5f;
    float fy0 = floorf(sy), fx0 = floorf(sx);
    float ty = sy - fy0, tx = sx - fx0;
    int y0 = (int)fy0, x0 = (int)fx0;
    int y1 = y0 + 1, x1 = x0 + 1;
    y0 = max(0, min(y0, S - 1)); y1 = max(0, min(y1, S - 1));
    x0 = max(0, min(x0, S - 1)); x1 = max(0, min(x1, S - 1));
    float v00 = img[y0 * S + x0], v01 = img[y0 * S + x1];
    float v10 = img[y1 * S + x0], v11 = img[y1 * S + x1];
    float top = fmaf(tx, v01 - v00, v00);
    float bot = fmaf(tx, v11 - v10, v10);
    return fmaf(ty, bot - top, top);
}

__global__ __launch_bounds__(256) void upsample_add(
    const float* __restrict__ f3p,  // [N, EMBED, 14,14]
    const float* __restrict__ f4p,  // [N, EMBED, 7,7]
    float* __restrict__ fmap,       // [N, EMBED, 28,28] (in/out)
    int total)                      // N*EMBED*784
{
    int idx = blockIdx.x * blockDim.x + threadIdx.x;
    if (idx >= total) return;
    const int w  = idx % 28;
    const int h  = (idx / 28) % 28;
    const int no = idx / HW2;       // n*EMBED + o
    float v = fmap[idx];
    v += bil(f3p + (size_t)no * HW3, 14, h, w);
    v += bil(f4p + (size_t)no * HW4, 7,  h, w);
    fmap[idx] = v;
}

// ---------------------------------------------------------------------------
// g[n,o] = mean over 784 spatial positions; one wave32 per (n,o) row.
// ---------------------------------------------------------------------------
__global__ __launch_bounds__(256) void reduce_g(
    const float* __restrict__ fmap, float* __restrict__ g, int totalRows)
{
    const int row  = blockIdx.x * 8 + (threadIdx.x >> 5);
    if (row >= totalRows) return;
    const int lane = threadIdx.x & 31;
    const float* __restrict__ r = fmap + (size_t)row * HW2;
    float s = 0.0f;
    for (int i = lane; i < HW2; i += 32) s += r[i];
    #pragma unroll
    for (int off = 16; off > 0; off >>= 1) s += __shfl_down(s, off, 32);
    if (lane == 0) g[row] = s * (1.0f / (float)HW2);
}

// ---------------------------------------------------------------------------
// heat[n,p] = mean over channels of |fmap[n,c,p]|
// ---------------------------------------------------------------------------
__global__ __launch_bounds__(256) void heat_kernel(
    const float* __restrict__ fmap, float* __restrict__ heat, int total)
{
    int idx = blockIdx.x * blockDim.x + threadIdx.x;
    if (idx >= total) return;
    const int n = idx / HW2;
    const int p = idx % HW2;
    const float* __restrict__ base = fmap + (size_t)n * EMBED * HW2 + p;
    float s = 0.0f;
    for (int c = 0; c < EMBED; ++c) s += fabsf(base[(size_t)c * HW2]);
    heat[idx] = s * (1.0f / (float)EMBED);
}

// ---------------------------------------------------------------------------
// Per-image top-8 argmax (lowest-index tie-break, matches lax.top_k)
// followed by channel-vector gather. tokens layout: [N, 8, EMBED]
// ---------------------------------------------------------------------------
__global__ __launch_bounds__(256) void topk_gather(
    const float* __restrict__ fmap,
    const float* __restrict__ heat,
    float* __restrict__ tokens)
{
    __shared__ float sh[HW2];
    __shared__ float sval[256];
    __shared__ int   sidx[256];
    __shared__ int   widx;

    const int n = blockIdx.x;
    const int t = threadIdx.x;

    for (int i = t; i < HW2; i += 256) sh[i] = heat[n * HW2 + i];
    __syncthreads();

    const float* __restrict__ fb = fmap + (size_t)n * EMBED * HW2;

    for (int k = 0; k < 8; ++k) {
        float bv = NEG_INF;
        int   bi = 0x7fffffff;
        for (int i = t; i < HW2; i += 256) {
            float v = sh[i];
            if (v > bv || (v == bv && i < bi)) { bv = v; bi = i; }
        }
        sval[t] = bv; sidx[t] = bi;
        __syncthreads();
        for (int s = 128; s > 0; s >>= 1) {
            if (t < s) {
                float ov = sval[t + s]; int oi = sidx[t + s];
                if (ov > sval[t] || (ov == sval[t] && oi < sidx[t])) {
                    sval[t] = ov; sidx[t] = oi;
                }
            }
            __syncthreads();
        }
        if (t == 0) { widx = sidx[0]; sh[sidx[0]] = NEG_INF; }
        __syncthreads();
        const int p = widx;
        for (int c = t; c < EMBED; c += 256) {
            tokens[((size_t)(n * 8 + k)) * EMBED + c] = fb[(size_t)c * HW2 + p];
        }
        __syncthreads();
    }
}

// ---------------------------------------------------------------------------
// Host-side launch
// ---------------------------------------------------------------------------
extern "C" void kernel_launch(void* const* d_in, const int* in_sizes, int n_in,
                              void* d_out, int out_size, void* d_ws, size_t ws_size,
                              hipStream_t stream)
{
    const float* f2 = (const float*)d_in[0];  // [128,192,28,28]
    const float* f3 = (const float*)d_in[1];  // [128,384,14,14]
    const float* f4 = (const float*)d_in[2];  // [128,768,7,7]
    const float* w2 = (const float*)d_in[3];  // [512,192]
    const float* b2 = (const float*)d_in[4];
    const float* w3 = (const float*)d_in[5];  // [512,384]
    const float* b3 = (const float*)d_in[6];
    const float* w4 = (const float*)d_in[7];  // [512,768]
    const float* b4 = (const float*)d_in[8];

    const int N = 128;  // B*T = 8*16

    float* ws   = (float*)d_ws;
    float* fmap = ws;                                    // N*512*784
    float* f3p  = fmap + (size_t)N * EMBED * HW2;        // N*512*196
    float* f4p  = f3p  + (size_t)N * EMBED * HW3;        // N*512*49
    float* heat = f4p  + (size_t)N * EMBED * HW4;        // N*784

    float* g_out      = (float*)d_out;                   // [N, 512]
    float* tokens_out = g_out + (size_t)N * EMBED;       // [N, 8, 512]

    const dim3 blk(256);

    // 1) projections (independent GEMMs), 32x32 wave tiles, 256 O-rows/block
    proj_gemm_wmma<192, HW2><<<dim3((HW2 + 31) / 32, 2, N), blk, 0, stream>>>(f2, w2, b2, fmap);
    proj_gemm_wmma<384, HW3><<<dim3((HW3 + 31) / 32, 2, N), blk, 0, stream>>>(f3, w3, b3, f3p);
    proj_gemm_wmma<768, HW4><<<dim3((HW4 + 31) / 32, 2, N), blk, 0, stream>>>(f4, w4, b4, f4p);

    // 2) residual cascade with bilinear upsampling
    {
        const int total = N * EMBED * HW2;
        upsample_add<<<dim3((total + 255) / 256), blk, 0, stream>>>(f3p, f4p, fmap, total);
    }

    // 3) global token (spatial mean)
    {
        const int rows = N * EMBED;
        reduce_g<<<dim3((rows + 7) / 8), blk, 0, stream>>>(fmap, g_out, rows);
    }

    // 4) heat map
    {
        const int total = N * HW2;
        heat_kernel<<<dim3((total + 255) / 256), blk, 0, stream>>>(fmap, heat, total);
    }

    // 5) top-8 + gather
    topk_gather<<<dim3(N), blk, 0, stream>>>(fmap, heat, tokens_out);
}